// NystromAttention_3040836845875
// MI455X (gfx1250) — compile-verified
//
#include <hip/hip_runtime.h>

typedef __attribute__((ext_vector_type(2))) float v2f;
typedef __attribute__((ext_vector_type(8))) float v8f;

#define WMMA_F32(a, b, c) \
  __builtin_amdgcn_wmma_f32_16x16x4_f32(false, (a), false, (b), (short)0, (c), false, false)

constexpr int HID  = 1024;
constexpr int D    = 64;
constexpr int NH   = 16;
constexpr int NL   = 128;
constexpr int LPAD = 4096;
constexpr int SLEN = 4095;
constexpr int BATCH = 2;
constexpr int SEG  = 32;
constexpr int NB   = BATCH * NH;          // 32 head-batches
constexpr float QK_SCALE = 0.35355339059327373f;  // D^-0.25 = 1/(2*sqrt(2))

// ===========================================================================
// Double-buffered WMMA GEMM core.
//  - A is row-major, aBase already points at row start + kh*2.
//  - TRANSB: bp[j] points at B row n_j (+kh*2): contiguous v2f loads.
//  - !TRANSB: bp[j] = B + n_j; elements at bp[j] + kk*N (two b32 loads).
//  - Chunk = 16 k-values = 4 WMMA k-steps; ping-pong register buffers so the
//    next chunk's loads are in flight while the current chunk's 4*NT WMMAs
//    execute (8 independent accumulator chains -> back-to-back XDL issue).
// ===========================================================================
template <bool TRANSB, int NT>
__device__ __forceinline__ void chunk_load(
    const float* __restrict__ aBase, float aMul,
    const float* const (&bp)[NT], long N, int kh2, int k0,
    v2f (&a)[4], v2f (&b)[NT][4])
{
#pragma unroll
  for (int s = 0; s < 4; ++s) {
    a[s] = *(const v2f*)(aBase + k0 + 4 * s);
    a[s] *= aMul;
  }
#pragma unroll
  for (int j = 0; j < NT; ++j) {
#pragma unroll
    for (int s = 0; s < 4; ++s) {
      if (TRANSB) {
        b[j][s] = *(const v2f*)(bp[j] + k0 + 4 * s);
      } else {
        const long kk = k0 + 4 * s + kh2;
        b[j][s].x = bp[j][kk * N];
        b[j][s].y = bp[j][(kk + 1) * N];
      }
    }
  }
}

template <int NT>
__device__ __forceinline__ void chunk_mma(const v2f (&a)[4], const v2f (&b)[NT][4], v8f* acc)
{
#pragma unroll
  for (int s = 0; s < 4; ++s) {
#pragma unroll
    for (int j = 0; j < NT; ++j) {
      acc[j] = WMMA_F32(a[s], b[j][s], acc[j]);
    }
  }
}

template <bool TRANSB, int NT>
__device__ __forceinline__ void gemm_loop(
    const float* __restrict__ aBase, float aMul,
    const float* const (&bp)[NT], long N, int kh2, int Kdim, v8f* acc)
{
  v2f aA[4], aB[4];
  v2f bA[NT][4], bB[NT][4];
  chunk_load<TRANSB, NT>(aBase, aMul, bp, N, kh2, 0, aA, bA);
  for (int k0 = 0; k0 < Kdim; k0 += 32) {
    if (k0 + 16 < Kdim) chunk_load<TRANSB, NT>(aBase, aMul, bp, N, kh2, k0 + 16, aB, bB);
    chunk_mma<NT>(aA, bA, acc);
    if (k0 + 32 < Kdim) chunk_load<TRANSB, NT>(aBase, aMul, bp, N, kh2, k0 + 32, aA, bA);
    if (k0 + 16 < Kdim) chunk_mma<NT>(aB, bB, acc);
  }
}

// ---------------------------------------------------------------------------
// QKV projection: C (8192 x 1024) = Xp @ W^T, epilogue bias (+ mask*scale),
// scatter to (B,NH,L,D).  Block = 128x128 tile, 8 waves x 16-row strips.
// ---------------------------------------------------------------------------
__global__ __launch_bounds__(256) void qkv_proj_kernel(
    const float* __restrict__ X, const float* __restrict__ W,
    const float* __restrict__ bias, const float* __restrict__ mask,
    float* __restrict__ out, int applyScale)
{
  const int wave = threadIdx.x >> 5;
  const int lane = threadIdx.x & 31;
  const int row  = lane & 15;
  const int kh2  = (lane >> 4) * 2;
  const int m0   = blockIdx.x * 128 + wave * 16;   // gridDim.x = 64
  const int n0   = blockIdx.y * 128;               // gridDim.y = 8

  const int m = m0 + row;
  const int b = m >> 12;
  const int l = m & (LPAD - 1);
  const float vA = (l < SLEN) ? 1.0f : 0.0f;
  const float* aBase = X + ((long)b * SLEN + (l < SLEN ? l : 0)) * HID + kh2;

  const float* bp[8];
#pragma unroll
  for (int j = 0; j < 8; ++j)
    bp[j] = W + (long)(n0 + j * 16 + (lane & 15)) * HID + kh2;

  v8f acc[8];
#pragma unroll
  for (int j = 0; j < 8; ++j) acc[j] = (v8f)0.0f;

  gemm_loop<true, 8>(aBase, vA, bp, HID, kh2, HID, acc);

#pragma unroll
  for (int j = 0; j < 8; ++j) {
    const int n = n0 + j * 16 + (lane & 15);
    const int h = n >> 6, d = n & 63;
    const float bs = bias[n];
#pragma unroll
    for (int r = 0; r < 8; ++r) {
      const int mm = m0 + (kh2 << 2) + r;          // lanes 16-31 -> M = r+8
      const int bb = mm >> 12, ll = mm & (LPAD - 1);
      float v = acc[j][r] + bs;
      if (applyScale) v *= mask[bb * LPAD + ll] * QK_SCALE;
      out[(((long)(bb * NH + h)) * LPAD + ll) * D + d] = v;
    }
  }
}

// ---------------------------------------------------------------------------
// Output projection: out = attn @ Wo^T + bo, drop pad rows (l == 4095)
// ---------------------------------------------------------------------------
__global__ __launch_bounds__(256) void out_proj_kernel(
    const float* __restrict__ Att, const float* __restrict__ Wo,
    const float* __restrict__ bo, float* __restrict__ out)
{
  const int wave = threadIdx.x >> 5;
  const int lane = threadIdx.x & 31;
  const int row  = lane & 15;
  const int kh2  = (lane >> 4) * 2;
  const int m0   = blockIdx.x * 128 + wave * 16;
  const int n0   = blockIdx.y * 128;

  const float* aBase = Att + (long)(m0 + row) * HID + kh2;
  const float* bp[8];
#pragma unroll
  for (int j = 0; j < 8; ++j)
    bp[j] = Wo + (long)(n0 + j * 16 + (lane & 15)) * HID + kh2;

  v8f acc[8];
#pragma unroll
  for (int j = 0; j < 8; ++j) acc[j] = (v8f)0.0f;

  gemm_loop<true, 8>(aBase, 1.0f, bp, HID, kh2, HID, acc);

#pragma unroll
  for (int j = 0; j < 8; ++j) {
    const int n = n0 + j * 16 + (lane & 15);
    const float bs = bo[n];
#pragma unroll
    for (int r = 0; r < 8; ++r) {
      const int mm = m0 + (kh2 << 2) + r;
      const int bb = mm >> 12, ll = mm & (LPAD - 1);
      if (ll < SLEN)
        out[((long)bb * SLEN + ll) * HID + n] = acc[j][r] + bs;
    }
  }
}

// ---------------------------------------------------------------------------
// Generic batched WMMA GEMM.  C1 = cscale*(A@Bop); C2 = alpha2*I - (A@Bop).
// Block computes a 128 x (NT*16) tile; M multiple of 128, lda == Kdim.
// ---------------------------------------------------------------------------
template <bool TRANSB, int NT>
__global__ __launch_bounds__(256) void bmm_kernel(
    const float* __restrict__ A, const float* __restrict__ Bm,
    float* __restrict__ C1, float* __restrict__ C2,
    int Kdim, int N, long aStride, long bStride,
    long ldc, int cDiv, long cOuter, long cInner,
    float cscale, float alpha2)
{
  const int wave = threadIdx.x >> 5;
  const int lane = threadIdx.x & 31;
  const int row  = lane & 15;
  const int kh2  = (lane >> 4) * 2;
  const int bh   = blockIdx.z;
  const int m0   = blockIdx.x * 128 + wave * 16;
  const int n0   = blockIdx.y * 128;

  const float* Ab = A + (long)bh * aStride;
  const float* Bb = Bm + (long)bh * bStride;

  const float* aBase = Ab + (long)(m0 + row) * Kdim + kh2;
  const float* bp[NT];
#pragma unroll
  for (int j = 0; j < NT; ++j) {
    const int n = n0 + j * 16 + (lane & 15);
    bp[j] = TRANSB ? (Bb + (long)n * Kdim + kh2) : (Bb + n);
  }

  v8f acc[NT];
#pragma unroll
  for (int j = 0; j < NT; ++j) acc[j] = (v8f)0.0f;

  gemm_loop<TRANSB, NT>(aBase, 1.0f, bp, N, kh2, Kdim, acc);

  const long cbase = (long)(bh / cDiv) * cOuter + (long)(bh % cDiv) * cInner;
#pragma unroll
  for (int j = 0; j < NT; ++j) {
    const int n = n0 + j * 16 + (lane & 15);
#pragma unroll
    for (int r = 0; r < 8; ++r) {
      const int mm = m0 + (kh2 << 2) + r;
      const float v = acc[j][r];
      if (C1) C1[cbase + (long)mm * ldc + n] = cscale * v;
      if (C2) C2[cbase + (long)mm * ldc + n] = ((mm == n) ? alpha2 : 0.0f) - v;
    }
  }
}

// ---------------------------------------------------------------------------
// Row softmax (in place), optional mask bias: v -= 1e9*(1 - mask[b, col])
// ---------------------------------------------------------------------------
__global__ __launch_bounds__(256) void rowsoftmax_kernel(
    float* __restrict__ buf, int cols, const float* __restrict__ mask,
    int rowsPerB)
{
  const long rowIdx = blockIdx.x;
  float* base = buf + rowIdx * (long)cols;
  const float* mrow = mask ? (mask + (rowIdx / rowsPerB) * LPAD) : nullptr;
  __shared__ float red[256];
  const int tid = threadIdx.x;

  float mx = -3.4e38f;
  for (int j = tid; j < cols; j += 256) {
    float v = base[j];
    if (mrow) v -= 1.0e9f * (1.0f - mrow[j]);
    mx = fmaxf(mx, v);
  }
  red[tid] = mx;
  __syncthreads();
  for (int s = 128; s > 0; s >>= 1) {
    if (tid < s) red[tid] = fmaxf(red[tid], red[tid + s]);
    __syncthreads();
  }
  mx = red[0];
  __syncthreads();

  float sum = 0.0f;
  for (int j = tid; j < cols; j += 256) {
    float v = base[j];
    if (mrow) v -= 1.0e9f * (1.0f - mrow[j]);
    sum += __expf(v - mx);
  }
  red[tid] = sum;
  __syncthreads();
  for (int s = 128; s > 0; s >>= 1) {
    if (tid < s) red[tid] += red[tid + s];
    __syncthreads();
  }
  const float inv = 1.0f / red[0];
  for (int j = tid; j < cols; j += 256) {
    float v = base[j];
    if (mrow) v -= 1.0e9f * (1.0f - mrow[j]);
    base[j] = __expf(v - mx) * inv;
  }
}

// ---------------------------------------------------------------------------
// Landmark means: dst[bh][n][d] = mean over SEG of src[bh][n*SEG+s][d]
// ---------------------------------------------------------------------------
__global__ __launch_bounds__(256) void landmark_kernel(
    const float* __restrict__ src, float* __restrict__ dst)
{
  const int idx = blockIdx.x * 256 + threadIdx.x;   // total NB*NL*D = 262144
  const int d = idx & 63;
  const int n = (idx >> 6) & (NL - 1);
  const int bh = idx >> 13;
  const float* p = src + ((long)bh * LPAD + n * SEG) * D + d;
  float s = 0.0f;
#pragma unroll
  for (int t = 0; t < SEG; ++t) s += p[t * D];
  dst[idx] = s * (1.0f / SEG);
}

// ---------------------------------------------------------------------------
// denom = max(colsum(|K|)) * max(rowsum(|K|)); V0 = K^T / denom  (per bh)
// ---------------------------------------------------------------------------
__global__ __launch_bounds__(256) void denom_v0_kernel(
    const float* __restrict__ Km, float* __restrict__ V0)
{
  const int bh = blockIdx.x;
  const float* Kb = Km + (long)bh * NL * NL;
  __shared__ float rs[NL], cs[NL];
  __shared__ float denom;
  const int tid = threadIdx.x;
  if (tid < NL) {
    float r = 0.0f, c = 0.0f;
    for (int j = 0; j < NL; ++j) {
      r += fabsf(Kb[tid * NL + j]);
      c += fabsf(Kb[j * NL + tid]);
    }
    rs[tid] = r; cs[tid] = c;
  }
  __syncthreads();
  if (tid == 0) {
    float mr = 0.0f, mc = 0.0f;
    for (int j = 0; j < NL; ++j) { mr = fmaxf(mr, rs[j]); mc = fmaxf(mc, cs[j]); }
    denom = mr * mc;
  }
  __syncthreads();
  const float invd = 1.0f / denom;
  for (int idx = tid; idx < NL * NL; idx += 256) {
    const int i = idx >> 7, j = idx & 127;
    V0[(long)bh * NL * NL + idx] = Kb[j * NL + i] * invd;
  }
}

// ---------------------------------------------------------------------------
extern "C" void kernel_launch(void* const* d_in, const int* /*in_sizes*/, int /*n_in*/,
                              void* d_out, int /*out_size*/, void* d_ws, size_t /*ws_size*/,
                              hipStream_t stream)
{
  const float* X    = (const float*)d_in[0];
  const float* mask = (const float*)d_in[1];
  const float* Wq   = (const float*)d_in[2];
  const float* bq   = (const float*)d_in[3];
  const float* Wk   = (const float*)d_in[4];
  const float* bk   = (const float*)d_in[5];
  const float* Wv   = (const float*)d_in[6];
  const float* bv   = (const float*)d_in[7];
  const float* Wo   = (const float*)d_in[8];
  const float* bo   = (const float*)d_in[9];

  float* ws = (float*)d_ws;
  size_t o = 0;
  const size_t SZ_QKV = (size_t)NB * LPAD * D;        // 8,388,608
  const size_t SZ_S   = (size_t)NB * NL * LPAD;       // 16,777,216
  const size_t SZ_L   = (size_t)NB * NL * D;          // 262,144
  const size_t SZ_M   = (size_t)NB * NL * NL;         // 524,288
  float* Q   = ws + o; o += SZ_QKV;
  float* K   = ws + o; o += SZ_QKV;                   // reused as attn later
  float* V   = ws + o; o += SZ_QKV;
  float* S   = ws + o; o += SZ_S;                     // S3 then S1
  float* Ql  = ws + o; o += SZ_L;
  float* Kl  = ws + o; o += SZ_L;
  float* Km  = ws + o; o += SZ_M;
  float* Va  = ws + o; o += SZ_M;
  float* Vb  = ws + o; o += SZ_M;
  float* KV  = ws + o; o += SZ_M;
  float* Ea  = ws + o; o += SZ_M;
  float* Eb  = ws + o; o += SZ_M;
  float* F3V = ws + o; o += SZ_L;
  float* T2  = ws + o; o += SZ_L;

  const dim3 blk(256);

  // 1) QKV projections (Q,K scaled by mask * D^-0.25)
  const dim3 gproj(64, 8, 1);
  qkv_proj_kernel<<<gproj, blk, 0, stream>>>(X, Wq, bq, mask, Q, 1);
  qkv_proj_kernel<<<gproj, blk, 0, stream>>>(X, Wk, bk, mask, K, 1);
  qkv_proj_kernel<<<gproj, blk, 0, stream>>>(X, Wv, bv, mask, V, 0);

  // 2) Landmarks
  landmark_kernel<<<1024, blk, 0, stream>>>(Q, Ql);
  landmark_kernel<<<1024, blk, 0, stream>>>(K, Kl);

  // 3) kernel_2 = softmax(Ql @ Kl^T)
  bmm_kernel<true, 8><<<dim3(1, 1, NB), blk, 0, stream>>>(
      Ql, Kl, Km, nullptr, D, NL, NL * D, NL * D,
      NL, NB, 0, (long)NL * NL, 1.0f, 0.0f);
  rowsoftmax_kernel<<<NB * NL, blk, 0, stream>>>(Km, NL, nullptr, 1);

  // 4) iterative_inv: V0 = K^T/denom; 6 Newton-Schulz iterations
  denom_v0_kernel<<<NB, blk, 0, stream>>>(Km, Va);
  float* Vc = Va;
  float* Vn = Vb;
  const long MM = (long)NL * NL;
  for (int it = 0; it < 6; ++it) {
    // KV = Km @ Vc ; Ea = 7I - KV
    bmm_kernel<false, 8><<<dim3(1, 1, NB), blk, 0, stream>>>(
        Km, Vc, KV, Ea, NL, NL, MM, MM, NL, NB, 0, MM, 1.0f, 7.0f);
    // Eb = 15I - KV @ Ea
    bmm_kernel<false, 8><<<dim3(1, 1, NB), blk, 0, stream>>>(
        KV, Ea, nullptr, Eb, NL, NL, MM, MM, NL, NB, 0, MM, 1.0f, 15.0f);
    // Ea = 13I - KV @ Eb
    bmm_kernel<false, 8><<<dim3(1, 1, NB), blk, 0, stream>>>(
        KV, Eb, nullptr, Ea, NL, NL, MM, MM, NL, NB, 0, MM, 1.0f, 13.0f);
    // Vn = 0.25 * Vc @ Ea
    bmm_kernel<false, 8><<<dim3(1, 1, NB), blk, 0, stream>>>(
        Vc, Ea, Vn, nullptr, NL, NL, MM, MM, NL, NB, 0, MM, 0.25f, 0.0f);
    float* t = Vc; Vc = Vn; Vn = t;
  }

  // 5) kernel_3 path: S = Ql @ K^T (128 x 4096), masked softmax, F3V = S @ V
  bmm_kernel<true, 8><<<dim3(1, 32, NB), blk, 0, stream>>>(
      Ql, K, S, nullptr, D, LPAD, NL * D, (long)LPAD * D,
      LPAD, NB, 0, (long)NL * LPAD, 1.0f, 0.0f);
  rowsoftmax_kernel<<<NB * NL, blk, 0, stream>>>(S, LPAD, mask, NH * NL);
  bmm_kernel<false, 4><<<dim3(1, 1, NB), blk, 0, stream>>>(
      S, V, F3V, nullptr, LPAD, D, (long)NL * LPAD, (long)LPAD * D,
      D, NB, 0, (long)NL * D, 1.0f, 0.0f);

  // 6) T2 = inv(kernel_2) @ F3V
  bmm_kernel<false, 4><<<dim3(1, 1, NB), blk, 0, stream>>>(
      Vc, F3V, T2, nullptr, NL, D, MM, (long)NL * D,
      D, NB, 0, (long)NL * D, 1.0f, 0.0f);

  // 7) kernel_1 path: S1 = Q @ Kl^T (4096 x 128), softmax, attn = S1 @ T2
  bmm_kernel<true, 8><<<dim3(32, 1, NB), blk, 0, stream>>>(
      Q, Kl, S, nullptr, D, NL, (long)LPAD * D, NL * D,
      NL, NB, 0, (long)LPAD * NL, 1.0f, 0.0f);
  rowsoftmax_kernel<<<NB * LPAD, blk, 0, stream>>>(S, NL, nullptr, 1);
  float* Att = K;  // K no longer needed; reuse as attn (B, L, NH*D)
  bmm_kernel<false, 4><<<dim3(32, 1, NB), blk, 0, stream>>>(
      S, T2, Att, nullptr, NL, D, (long)LPAD * NL, (long)NL * D,
      (long)NH * D, NH, (long)LPAD * NH * D, D, 1.0f, 0.0f);

  // 8) Output projection + bias, drop pad rows
  out_proj_kernel<<<dim3(64, 8, 1), blk, 0, stream>>>(Att, Wo, bo, (float*)d_out);
}